// YoloOutputDecoderLayer_71957882077671
// MI455X (gfx1250) — compile-verified
//
#include <hip/hip_runtime.h>
#include <cstdint>
#include <cstddef>

// YOLO 1D (audio) detection decode for MI455X (gfx1250).
//
// Memory-bound: 128 MiB in + 128 MiB out @ 23.3 TB/s HBM ~= 11.5 us.
// No matmul structure -> WMMA is not applicable; the CDNA5 feature this
// kernel rewards is the async LDS->global store path (ASYNCcnt-tracked
// GLOBAL_STORE_ASYNC_FROM_LDS_B128) to solve the 28-byte-row coalescing
// problem, plus non-temporal b128 loads for streaming reads.

typedef float vfloat4 __attribute__((ext_vector_type(4)));

#define TPB 256
#define CONF_THRESH 0.5f
#define GRID_CELLS 131072          // G
#define CELL_COEF 8.0f             // INPUT_LENGTH / G = 1048576 / 131072
#define LEN_COEF 1048576.0f        // INPUT_LENGTH

__global__ __launch_bounds__(TPB)
void yolo_decode_kernel(const vfloat4* __restrict__ in,
                        float* __restrict__ frag,
                        float* __restrict__ maskf,
                        int ncells)
{
    // Staging tile: 256 cells * 7 floats = 7168 B (contiguous output slice
    // of this block). Stride-7 per-lane LDS writes are bank-conflict-free
    // (gcd(7,64)=1).
    __shared__ float smem[TPB * 7];

    const int tid = threadIdx.x;
    const long long block0 = (long long)blockIdx.x * TPB;
    const long long cell  = block0 + tid;

    if (cell < ncells) {
        // 32 B/cell as two NT b128 loads (streaming; don't pollute L2).
        vfloat4 a = __builtin_nontemporal_load(in + cell * 2);
        vfloat4 b = __builtin_nontemporal_load(in + cell * 2 + 1);

        const float g = (float)((int)(cell & (GRID_CELLS - 1)));
        const bool  m = a.x >= CONF_THRESH;

        const float center = (g + a.y) * CELL_COEF;
        const float halfd  = a.z * (0.5f * LEN_COEF);
        // jnp.round == round-half-to-even == rintf under default RNE mode
        const float s = rintf(center - halfd);
        const float e = rintf(center + halfd);

        float* row = &smem[tid * 7];
        row[0] = m ? s   : 0.0f;
        row[1] = m ? e   : 0.0f;
        row[2] = m ? a.w : 0.0f;
        row[3] = m ? b.x : 0.0f;
        row[4] = m ? b.y : 0.0f;
        row[5] = m ? b.z : 0.0f;
        row[6] = m ? b.w : 0.0f;

        // Mask lane-coalesced directly (b32 NT store).
        __builtin_nontemporal_store(m ? 1.0f : 0.0f, maskf + cell);
    }
    __syncthreads();   // staging visible before async engine reads LDS

    if (block0 + TPB <= ncells) {
        // Full tile: stream 7168 B LDS -> global as 448 aligned 16B chunks
        // using the CDNA5 async store path (ASYNCcnt). Each enabled lane
        // moves one b128 chunk per instruction; 32 lanes = 512 B/issue,
        // no VGPR round-trip for the payload.
        const float* gbase = frag + block0 * 7;   // blockIdx*7168 B, 16B-aligned
        #pragma unroll
        for (int c = tid; c < (TPB * 7) / 4; c += TPB) {
            // Low 32 bits of a generic LDS pointer == wave-relative LDS
            // byte offset (ISA 10.2).
            unsigned lds = (unsigned)(size_t)(&smem[c * 4]);
            unsigned long long gaddr =
                (unsigned long long)(size_t)(gbase + c * 4);
            asm volatile("global_store_async_from_lds_b128 %0, %1, off"
                         :: "v"(gaddr), "v"(lds)
                         : "memory");
        }
        // Drain ASYNCcnt before wave teardown / LDS reuse.
        asm volatile("s_wait_asynccnt 0x0" ::: "memory");
    } else if (cell < ncells) {
        // Generic tail (never taken for this workload: 4,194,304 % 256 == 0).
        float* row = frag + cell * 7;
        const float* srow = &smem[tid * 7];
        #pragma unroll
        for (int j = 0; j < 7; ++j) row[j] = srow[j];
    }
}

extern "C" void kernel_launch(void* const* d_in, const int* in_sizes, int n_in,
                              void* d_out, int out_size, void* d_ws, size_t ws_size,
                              hipStream_t stream) {
    (void)n_in; (void)out_size; (void)d_ws; (void)ws_size;

    const int ncells = in_sizes[0] / 8;                 // B*G = 4,194,304
    const vfloat4* in = (const vfloat4*)d_in[0];
    float* frag  = (float*)d_out;                       // [B*G, 7]
    float* maskf = frag + (size_t)ncells * 7;           // [B*G] as float 0/1

    const int blocks = (ncells + TPB - 1) / TPB;        // 16384
    yolo_decode_kernel<<<blocks, TPB, 0, stream>>>(in, frag, maskf, ncells);
}